// SpikingDense_71476845740372
// MI455X (gfx1250) — compile-verified
//
#include <hip/hip_runtime.h>

typedef float v2f __attribute__((ext_vector_type(2)));
typedef float v8f __attribute__((ext_vector_type(8)));

#define BATCH 32
#define DIN   1024
#define UOUT  1024
#define FARF  1.0e6f

__global__ __launch_bounds__(256) void spiking_dense_wmma_kernel(
    const float* __restrict__ tj,     // [B, DIN]
    const float* __restrict__ Kmat,   // [DIN, UOUT]
    const float* __restrict__ alpha,  // [UOUT]
    const float* __restrict__ thr,    // [UOUT]
    const float* __restrict__ tminp,  // scalar
    const float* __restrict__ tmaxp,  // scalar
    float* __restrict__ out)          // [B, UOUT]
{
    __shared__ float svals[DIN];   // sorted spike times
    __shared__ int   sidx[DIN];    // argsort order

    const int b   = blockIdx.x;
    const int tid = threadIdx.x;

    // ---- Stage 1: load + bitonic sort (stable via index tie-break) ----
    for (int i = tid; i < DIN; i += 256) {
        svals[i] = tj[b * DIN + i];
        sidx[i]  = i;
    }
    __syncthreads();

    for (int k = 2; k <= DIN; k <<= 1) {
        for (int j = k >> 1; j > 0; j >>= 1) {
            for (int t = tid; t < DIN; t += 256) {
                int ixj = t ^ j;
                if (ixj > t) {
                    float v0 = svals[t], v1 = svals[ixj];
                    int   a0 = sidx[t],  a1 = sidx[ixj];
                    bool  gt = (v0 > v1) || ((v0 == v1) && (a0 > a1));
                    bool  up = ((t & k) == 0);
                    if (gt == up) {
                        svals[t] = v1; svals[ixj] = v0;
                        sidx[t]  = a1; sidx[ixj]  = a0;
                    }
                }
            }
            __syncthreads();
        }
    }

    // ---- Stage 2: per-wave WMMA prefix-sum scan over sorted positions ----
    const int lane = tid & 31;
    const int wave = tid >> 5;
    const int n    = lane & 15;     // column within 16-unit tile
    const int half = lane >> 4;     // 0: rows 0-7 of block, 1: rows 8-15
    const int u    = blockIdx.y * 128 + wave * 16 + n;

    const float alphaU = alpha[u];
    const float addN   = thr[u] + alphaU * tminp[0];
    const float tmax0  = tmaxp[0];

    const float* Kcol = Kmat + u;   // row stride = UOUT

    float Sprev = 0.f, Tprev = 0.f;       // running cumsum carries
    float best_num = 0.f, best_den = 1.f; // selected numerator/denominator
    int   found = 0;

    for (int blk = 0; blk < DIN / 16; ++blk) {
        const int i0 = blk * 16;

        v8f cS = {0.f, 0.f, 0.f, 0.f, 0.f, 0.f, 0.f, 0.f};
        v8f cT = {0.f, 0.f, 0.f, 0.f, 0.f, 0.f, 0.f, 0.f};

        // P = L(16x16 lower-tri ones) x Jblk via 4 chained 16x16x4 f32 WMMAs,
        // done twice (J and J*t) -> within-block prefix sums for 16 units.
        #pragma unroll
        for (int c = 0; c < 4; ++c) {
            const int kg = 4 * c + 2 * half;     // global K index of this lane's pair
            const int ir = i0 + kg;
            const int row0 = sidx[ir];
            const int row1 = sidx[ir + 1];
            const float j0 = Kcol[(size_t)row0 * UOUT];
            const float j1 = Kcol[(size_t)row1 * UOUT];
            const float t0 = svals[ir];
            const float t1 = svals[ir + 1];

            v2f a, bS, bT;
            a[0] = (kg     <= n) ? 1.f : 0.f;    // triangular A, lane = M = n
            a[1] = (kg + 1 <= n) ? 1.f : 0.f;
            bS[0] = j0;      bS[1] = j1;
            bT[0] = j0 * t0; bT[1] = j1 * t1;

            cS = __builtin_amdgcn_wmma_f32_16x16x4_f32(
                false, a, false, bS, (short)0, cS, false, false);
            cT = __builtin_amdgcn_wmma_f32_16x16x4_f32(
                false, a, false, bT, (short)0, cT, false, false);
        }

        // Per-lane scan of its 8 rows (m = half*8 + r), first-hit detection.
        int   loc_hit = 8;
        float loc_num = 0.f, loc_den = 0.f;
        float num_r0 = 0.f, den_r0 = 0.f;
        #pragma unroll
        for (int r = 0; r < 8; ++r) {
            const int   i   = i0 + half * 8 + r;
            const float den = Sprev + cS[r] + alphaU;
            const float num = Tprev + cT[r] + addN;
            const float tjn = (i < DIN - 1) ? svals[i + 1] : FARF;
            const int   hit = (den > 0.f) && (num < tjn * den);
            const int   take = hit & (loc_hit == 8);
            loc_hit = take ? r   : loc_hit;
            loc_num = take ? num : loc_num;
            loc_den = take ? den : loc_den;
            if (r == 0) { num_r0 = num; den_r0 = den; }
        }

        // Merge low half (rows 0-7, priority) with high half (rows 8-15).
        const int   o_hit = __shfl_xor(loc_hit, 16, 32);
        const float o_num = __shfl_xor(loc_num, 16, 32);
        const float o_den = __shfl_xor(loc_den, 16, 32);
        const int   lo_hit = half ? o_hit : loc_hit;
        const float lo_num = half ? o_num : loc_num;
        const float lo_den = half ? o_den : loc_den;
        const int   hi_hit = half ? loc_hit : o_hit;
        const float hi_num = half ? loc_num : o_num;
        const float hi_den = half ? loc_den : o_den;
        const int   m_has = (lo_hit < 8) | (hi_hit < 8);
        const float m_num = (lo_hit < 8) ? lo_num : hi_num;
        const float m_den = (lo_hit < 8) ? lo_den : hi_den;

        if (blk == 0) {
            // argmax fallback: ti at sorted position 0 (held by low lane n)
            best_num = __shfl(num_r0, n, 32);
            best_den = __shfl(den_r0, n, 32);
        }

        const int take = (!found) & m_has;
        best_num = take ? m_num : best_num;
        best_den = take ? m_den : best_den;
        found |= m_has;

        // Carry = block totals = prefix at row 15 (high lane, VGPR 7).
        Sprev += __shfl(cS[7], n + 16, 32);
        Tprev += __shfl(cT[7], n + 16, 32);

        if (__all(found)) break;   // wave-uniform early exit
    }

    const float ti  = best_num / best_den;
    const float res = (ti <= tmax0) ? ti : tmax0;   // NaN -> tmax, as in reference
    out[b * UOUT + u] = res;
}

extern "C" void kernel_launch(void* const* d_in, const int* in_sizes, int n_in,
                              void* d_out, int out_size, void* d_ws, size_t ws_size,
                              hipStream_t stream) {
    (void)in_sizes; (void)n_in; (void)out_size; (void)d_ws; (void)ws_size;
    const float* tj    = (const float*)d_in[0];
    const float* Kmat  = (const float*)d_in[1];
    const float* alpha = (const float*)d_in[2];
    const float* thr   = (const float*)d_in[3];
    const float* tminp = (const float*)d_in[4];
    const float* tmaxp = (const float*)d_in[5];
    float* out = (float*)d_out;

    dim3 grid(BATCH, UOUT / 128);   // 32 x 8 workgroups
    dim3 block(256);                // 8 waves: sort cooperatively, scan 16 units/wave
    hipLaunchKernelGGL(spiking_dense_wmma_kernel, grid, block, 0, stream,
                       tj, Kmat, alpha, thr, tminp, tmaxp, out);
}